// MultimodalKroneckerFusion_81492709474484
// MI455X (gfx1250) — compile-verified
//
#include <hip/hip_runtime.h>
#include <hip/hip_bf16.h>

typedef __attribute__((ext_vector_type(2))) float v2f;
typedef __attribute__((ext_vector_type(8))) float v8f;
typedef __attribute__((ext_vector_type(4))) int   v4i;

#define AS_GLOBAL __attribute__((address_space(1)))
#define AS_LOCAL  __attribute__((address_space(3)))

// Async global->LDS copy (CDNA5, tracked by ASYNCcnt). The builtin exists on
// this toolchain and takes (global int4*, local int4*, imm offset, imm cpol).
#if __has_builtin(__builtin_amdgcn_global_load_async_to_lds_b128)
#define ASYNC_COPY_B128(gsrc, ldst)                                          \
    __builtin_amdgcn_global_load_async_to_lds_b128(                          \
        (AS_GLOBAL v4i*)(unsigned long long)(gsrc),                          \
        (AS_LOCAL  v4i*)(unsigned)(unsigned long long)(ldst), 0, 0)
#else
#define ASYNC_COPY_B128(gsrc, ldst)                                          \
    asm volatile("global_load_async_to_lds_b128 %0, %1, off"                 \
                 :: "v"((unsigned)(unsigned long long)(ldst)),               \
                    "v"((unsigned long long)(gsrc))                          \
                 : "memory")
#endif

#if __has_builtin(__builtin_amdgcn_s_wait_asynccnt)
#define WAIT_ASYNC0() __builtin_amdgcn_s_wait_asynccnt(0)
#else
#define WAIT_ASYNC0() asm volatile("s_wait_asynccnt 0" ::: "memory")
#endif

// ---------------------------------------------------------------------------
// Zero-fill (needed before split-K atomic accumulation; graph-capture safe)
// ---------------------------------------------------------------------------
__global__ void zero_f32_kernel(float* __restrict__ p, int n) {
    int i = blockIdx.x * blockDim.x + threadIdx.x;
    if (i < n) p[i] = 0.0f;
}

// ---------------------------------------------------------------------------
// On-the-fly Kronecker GEMM:  Z[b,o] += sum_k (x1[b,k>>LG_D2]*x2[b,k&(D2-1)]) * W[o,k]
// over K slice [blockIdx.y*sliceK, ...+sliceK).
//
// Block = 256 threads = 8 waves; block tile = M=256 x N=64.
//   wave w -> rows [32w, 32w+32): 2 M-subtiles x 4 N-subtiles of 16x16.
// W is staged per-block into LDS in 64-row x 128-col chunks with
// GLOBAL_LOAD_ASYNC_TO_LDS_B128 (double-buffered, ASYNCcnt + 1 barrier/chunk):
// each W element is fetched from global exactly once per block and B fragments
// come from conflict-free ds_load (stride 132 dwords -> banks (4r+2h+kk)%64
// all distinct across the 32 lanes).
// x2 fragments are register software-pipelined 4 K-steps (16 cols) ahead so
// their global-load latency is covered by ~32 WMMAs instead of ~8.
//
// f32 16x16x4 lane layout (ISA 7.12.2):
//   A (16x4): lane r / r+16 hold row M=r; VGPR0=K{0|2}, VGPR1=K{1|3}
//   B (4x16): lane n / n+16 hold col N=n; VGPR0=K{0|2}, VGPR1=K{1|3}
//   C/D     : VGPR t -> M = t + 8*(lane>>4), N = lane&15
// ---------------------------------------------------------------------------
#define KB      128          // K columns per LDS chunk
#define LDS_W   (KB + 4)     // +4 dword pad: stride 132 -> bank-conflict-free

template <int LG_D2>
__global__ __launch_bounds__(256) void kron_gemm_kernel(
    const float* __restrict__ X1,   // [256, d1]
    const float* __restrict__ X2,   // [256, D2]
    const float* __restrict__ W,    // [Nout, Ktot] row-major (torch Linear)
    float* __restrict__ Z,          // [256, Nout] fp32, atomically accumulated
    int d1, int Nout, long long Ktot, int sliceK)
{
    constexpr int D2 = 1 << LG_D2;
    __shared__ float ldsW[2][64][LDS_W];

    const int tid  = threadIdx.x;
    const int lane = tid & 31;
    const int wave = tid >> 5;
    const int r    = lane & 15;     // A row / B col within 16-tile
    const int half = lane >> 4;     // selects K pair {0,1} vs {2,3}

    const int mBase = wave * 32;               // block covers all M=256
    const int nBase = blockIdx.x * 64;         // N tile
    const long long k0 = (long long)blockIdx.y * (long long)sliceK;
    const int nChunks = sliceK / KB;

    // Per-lane x2 / x1 row pointers for the 2 M-subtiles.
    const float* x2p[2];
    const float* x1p[2];
#pragma unroll
    for (int mt = 0; mt < 2; ++mt) {
        const int row = mBase + mt * 16 + r;
        x2p[mt] = X2 + (size_t)row * D2 + 2 * half;
        x1p[mt] = X1 + (size_t)row * d1;
    }

    // Cooperative async copy of one 64 x KB W chunk into ldsW[buf].
    // 64*128 floats = 2048 float4; 256 threads x 8 transfers, coalesced.
    auto issue_chunk = [&](int c) {
        const long long kc = k0 + (long long)c * KB;
        const int buf = c & 1;
#pragma unroll
        for (int it = 0; it < 8; ++it) {
            const int f   = tid + it * 256;   // 0..2047
            const int row = f >> 5;           // 32 float4 per row
            const int c4  = (f & 31) << 2;    // float offset within row
            const float* gsrc = W + (size_t)(nBase + row) * (size_t)Ktot
                                  + (size_t)kc + c4;
            const float* ldst = &ldsW[buf][row][c4];
            ASYNC_COPY_B128(gsrc, ldst);
        }
    };

    v8f acc[2][4] = {};

    issue_chunk(0);

    for (int c = 0; c < nChunks; ++c) {
        // copy(c) is the only outstanding async op for this wave
        WAIT_ASYNC0();
        __syncthreads();                      // chunk c visible; buf[(c+1)&1] free
        if (c + 1 < nChunks) issue_chunk(c + 1);

        const long long kc = k0 + (long long)c * KB;
        const int i  = (int)(kc >> LG_D2);    // chunk lies within one i-run
        const int jb = (int)(kc & (D2 - 1));
        const float s0 = x1p[0][i];
        const float s1 = x1p[1][i];

        const int buf = c & 1;
        const float* bp0 = &ldsW[buf][ 0 + r][2 * half];
        const float* bp1 = &ldsW[buf][16 + r][2 * half];
        const float* bp2 = &ldsW[buf][32 + r][2 * half];
        const float* bp3 = &ldsW[buf][48 + r][2 * half];

        // 8 WMMAs for the K=4 step at kk, A scaled from the pipelined x2 pair
        auto compute4 = [&](int kk, float2 xa, float2 xb) {
            v2f a0; a0[0] = s0 * xa.x; a0[1] = s0 * xa.y;
            v2f a1; a1[0] = s1 * xb.x; a1[1] = s1 * xb.y;

            const float2 w0 = *reinterpret_cast<const float2*>(bp0 + kk);
            const float2 w1 = *reinterpret_cast<const float2*>(bp1 + kk);
            const float2 w2 = *reinterpret_cast<const float2*>(bp2 + kk);
            const float2 w3 = *reinterpret_cast<const float2*>(bp3 + kk);
            v2f b0; b0[0] = w0.x; b0[1] = w0.y;
            v2f b1; b1[0] = w1.x; b1[1] = w1.y;
            v2f b2; b2[0] = w2.x; b2[1] = w2.y;
            v2f b3; b3[0] = w3.x; b3[1] = w3.y;

            acc[0][0] = __builtin_amdgcn_wmma_f32_16x16x4_f32(false, a0, false, b0, (short)0, acc[0][0], false, false);
            acc[1][0] = __builtin_amdgcn_wmma_f32_16x16x4_f32(false, a1, false, b0, (short)0, acc[1][0], false, false);
            acc[0][1] = __builtin_amdgcn_wmma_f32_16x16x4_f32(false, a0, false, b1, (short)0, acc[0][1], false, false);
            acc[1][1] = __builtin_amdgcn_wmma_f32_16x16x4_f32(false, a1, false, b1, (short)0, acc[1][1], false, false);
            acc[0][2] = __builtin_amdgcn_wmma_f32_16x16x4_f32(false, a0, false, b2, (short)0, acc[0][2], false, false);
            acc[1][2] = __builtin_amdgcn_wmma_f32_16x16x4_f32(false, a1, false, b2, (short)0, acc[1][2], false, false);
            acc[0][3] = __builtin_amdgcn_wmma_f32_16x16x4_f32(false, a0, false, b3, (short)0, acc[0][3], false, false);
            acc[1][3] = __builtin_amdgcn_wmma_f32_16x16x4_f32(false, a1, false, b3, (short)0, acc[1][3], false, false);
        };

        // 4-deep register pipeline of x2 pairs (16 columns of lookahead)
        float2 fa[4], fb[4];
#pragma unroll
        for (int p = 0; p < 4; ++p) {
            fa[p] = *reinterpret_cast<const float2*>(x2p[0] + jb + 4 * p);
            fb[p] = *reinterpret_cast<const float2*>(x2p[1] + jb + 4 * p);
        }

#pragma unroll 4
        for (int kk = 0; kk < KB - 16; kk += 4) {
            const int slot = (kk >> 2) & 3;   // compile-time under unroll-4
            const float2 xa = fa[slot];
            const float2 xb = fb[slot];
            fa[slot] = *reinterpret_cast<const float2*>(x2p[0] + jb + kk + 16);
            fb[slot] = *reinterpret_cast<const float2*>(x2p[1] + jb + kk + 16);
            compute4(kk, xa, xb);
        }
#pragma unroll
        for (int kk = KB - 16; kk < KB; kk += 4) {
            const int slot = (kk >> 2) & 3;
            compute4(kk, fa[slot], fb[slot]);
        }
    }

    // Split-K partial accumulation into Z (fp32 atomics).
#pragma unroll
    for (int mt = 0; mt < 2; ++mt)
#pragma unroll
        for (int nt = 0; nt < 4; ++nt)
#pragma unroll
            for (int t = 0; t < 8; ++t) {
                const int row = mBase + mt * 16 + t + 8 * half;
                const int col = nBase + nt * 16 + r;
                atomicAdd(&Z[(size_t)row * Nout + col], acc[mt][nt][t]);
            }
}

// ---------------------------------------------------------------------------
// Epilogue: h = ReLU( (z + bias - mu) * rsqrt(var + eps) * gamma + beta )
// One block per batch row. wave32 shuffle + LDS reduction.
// ---------------------------------------------------------------------------
__global__ __launch_bounds__(256) void ln_relu_kernel(
    const float* __restrict__ Z, const float* __restrict__ bias,
    const float* __restrict__ gamma, const float* __restrict__ beta,
    float* __restrict__ Out, int O)
{
    const int row = blockIdx.x;
    const float* z = Z + (size_t)row * O;
    float* out = Out + (size_t)row * O;

    float sum = 0.0f, sumsq = 0.0f;
    for (int c = threadIdx.x; c < O; c += blockDim.x) {
        const float v = z[c] + bias[c];
        sum += v;
        sumsq += v * v;
    }

#pragma unroll
    for (int off = 16; off > 0; off >>= 1) {
        sum   += __shfl_xor(sum, off, 32);
        sumsq += __shfl_xor(sumsq, off, 32);
    }

    __shared__ float s1[8], s2[8];
    const int wave = threadIdx.x >> 5;
    const int lane = threadIdx.x & 31;
    if (lane == 0) { s1[wave] = sum; s2[wave] = sumsq; }
    __syncthreads();

    if (wave == 0) {
        float a = (lane < 8) ? s1[lane] : 0.0f;
        float b = (lane < 8) ? s2[lane] : 0.0f;
#pragma unroll
        for (int off = 16; off > 0; off >>= 1) {
            a += __shfl_xor(a, off, 32);
            b += __shfl_xor(b, off, 32);
        }
        if (lane == 0) { s1[0] = a; s2[0] = b; }
    }
    __syncthreads();

    const float invO = 1.0f / (float)O;
    const float mu   = s1[0] * invO;
    const float var  = s2[0] * invO - mu * mu;
    const float rstd = rsqrtf(var + 1e-5f);

    for (int c = threadIdx.x; c < O; c += blockDim.x) {
        const float v = z[c] + bias[c];
        const float h = (v - mu) * rstd * gamma[c] + beta[c];
        out[c] = fmaxf(h, 0.0f);
    }
}

// ---------------------------------------------------------------------------
// Launch: stage1 GEMM -> LN/ReLU -> stage2 GEMM -> LN/ReLU
// ---------------------------------------------------------------------------
extern "C" void kernel_launch(void* const* d_in, const int* in_sizes, int n_in,
                              void* d_out, int out_size, void* d_ws, size_t ws_size,
                              hipStream_t stream) {
    (void)in_sizes; (void)n_in; (void)out_size; (void)ws_size;

    const float* x1  = (const float*)d_in[0];   // [256, 512]
    const float* x2  = (const float*)d_in[1];   // [256, 512]
    const float* x3  = (const float*)d_in[2];   // [256, 256]
    const float* W1  = (const float*)d_in[3];   // [512, 262144]
    const float* b1  = (const float*)d_in[4];
    const float* g1  = (const float*)d_in[5];
    const float* be1 = (const float*)d_in[6];
    const float* W2  = (const float*)d_in[7];   // [256, 131072]
    const float* b2  = (const float*)d_in[8];
    const float* g2  = (const float*)d_in[9];
    const float* be2 = (const float*)d_in[10];
    float* out = (float*)d_out;                 // [256, 256]

    constexpr int B = 256;
    constexpr int D1 = 512, D2 = 512, D3 = 256;
    constexpr int O1 = 512, O2 = 256;

    float* z1    = (float*)d_ws;                 // 256*512
    float* z2    = z1 + (size_t)B * O1;          // 256*256
    float* fused = z2 + (size_t)B * O2;          // 256*512

    // 1) zero the split-K accumulators (z1, z2 contiguous)
    {
        const int n = B * O1 + B * O2;
        zero_f32_kernel<<<(n + 255) / 256, 256, 0, stream>>>(z1, n);
    }

    // 2) stage 1 GEMM: M=256, N=512, K=262144. sliceK=8192 (64 chunks), 32 slices.
    kron_gemm_kernel<9><<<dim3(O1 / 64, 32), 256, 0, stream>>>(
        x1, x2, W1, z1, D1, O1, (long long)D1 * D2, 8192);

    // 3) bias + LayerNorm(512) + ReLU -> fused
    ln_relu_kernel<<<B, 256, 0, stream>>>(z1, b1, g1, be1, fused, O1);

    // 4) stage 2 GEMM: M=256, N=256, K=131072. sliceK=2048 (16 chunks), 64 slices.
    kron_gemm_kernel<8><<<dim3(O2 / 64, 64), 256, 0, stream>>>(
        fused, x3, W2, z2, O1, O2, (long long)O1 * D3, 2048);

    // 5) bias + LayerNorm(256) + ReLU -> output
    ln_relu_kernel<<<B, 256, 0, stream>>>(z2, b2, g2, be2, out, O2);
}